// EquivariantLayer_11081015623805
// MI455X (gfx1250) — compile-verified
//
#include <hip/hip_runtime.h>

// ---------------------------------------------------------------------------
// CDNA5 (gfx1250) implementation: all FFT stages expressed as f32 WMMA GEMMs
// against precomputed DFT basis matrices (V_WMMA_F32_16X16X4_F32).
// wave32, 256 threads/block = 8 waves, one 16x16 C-tile per wave-task.
// Field staging uses GLOBAL_LOAD_ASYNC_TO_LDS_B32 (ASYNCcnt path).
// ---------------------------------------------------------------------------

typedef float v2f __attribute__((ext_vector_type(2)));
typedef float v8f __attribute__((ext_vector_type(8)));

#define PI2F 6.28318530717958647692f

// ---- workspace layout (float offsets) -------------------------------------
#define TAB_COSA   0          // [64 x 48]  cos(2pi x ky/64), ky>=33 -> 0
#define TAB_NSINA  3072       // [64 x 48] -sin
#define TAB_CF     6144       // [64 x 64]  cos(2pi kx y/64)
#define TAB_SF     10240      // [64 x 64]  sin
#define TAB_NSF    14336      // [64 x 64] -sin
#define TAB_E1C    18432      // [128 x 64] cos(2pi kx_signed y/128)
#define TAB_E1S    26624      // [128 x 64] sin
#define TAB_NE1S   34816      // [128 x 64] -sin
#define TAB_W2C    43008      // [48 x 128] w*cos(2pi ky x/128)/16384, ky>=33->0
#define TAB_W2NS   49152      // [48 x 128] -w*sin(...)/16384
#define TAB_TOTAL  55296

#define WS_REFF    55296      // [128 fld][64x33] Re rfft2(f)
#define WS_IMFF    325632
#define WS_KSYM    595968     // [128 fld][64x64] symmetrized kernel
#define WS_REKS    1120256    // [128 fld][64x33] Re rfft2(ksym)
#define WS_IMKS    1390592
#define WS_RECONV  1660928    // [256 fld][64x33]
#define WS_IMCONV  2201600
#define WS_U       2742272    // [256 fld][128x128]
#define WS_V       6936576

// ---- f32 WMMA 16x16x4 micro-kernel ----------------------------------------
__device__ __forceinline__ v8f wmma4(v2f a, v2f b, v8f c) {
  // D = A(16x4) * B(4x16) + C(16x16), f32
  return __builtin_amdgcn_wmma_f32_16x16x4_f32(false, a, false, b, (short)0, c,
                                               false, false);
}

__device__ __forceinline__ v8f vzero8() {
  v8f z = {0.f, 0.f, 0.f, 0.f, 0.f, 0.f, 0.f, 0.f};
  return z;
}

// Accumulate A(16xK, per-lane row pointer) * B(Kx16, per-lane col pointer).
// Arow = &A[row(lane)][0] (contiguous K), Bcol = &B[0][col(lane)] (stride LDB).
template <int K, int LDB>
__device__ __forceinline__ v8f mm_chain(const float* __restrict__ Arow,
                                        const float* __restrict__ Bcol,
                                        int half, v8f acc) {
#pragma unroll
  for (int kk = 0; kk < K; kk += 4) {
    int k0 = kk + 2 * half;
    v2f a, b;
    a.x = Arow[k0];
    a.y = Arow[k0 + 1];
    b.x = Bcol[(long long)k0 * LDB];
    b.y = Bcol[(long long)(k0 + 1) * LDB];
    acc = wmma4(a, b, acc);
  }
  return acc;
}

// ---------------------------------------------------------------------------
// Kernel 0: build DFT basis tables (exact modular phase reduction).
// ---------------------------------------------------------------------------
__global__ void __launch_bounds__(256) build_tables_kernel(float* __restrict__ t) {
  int g = blockIdx.x * 256 + threadIdx.x;
  if (g < 6144) {                       // cosA / nsinA  [64 x 48]
    int which = g / 3072, i = g % 3072;
    int x = i / 48, ky = i % 48;
    float v = 0.f;
    if (ky < 33) {
      float th = PI2F * (float)((x * ky) & 63) * (1.0f / 64.0f);
      v = which ? -sinf(th) : cosf(th);
    }
    t[g] = v;
  } else if (g < 18432) {               // Cf / Sf / NSf  [64 x 64]
    int i = g - 6144;
    int which = i / 4096, e = i % 4096;
    int kx = e / 64, y = e % 64;
    float th = PI2F * (float)((kx * y) & 63) * (1.0f / 64.0f);
    t[g] = (which == 0) ? cosf(th) : (which == 1 ? sinf(th) : -sinf(th));
  } else if (g < 43008) {               // E1c / E1s / NE1s  [128 x 64]
    int i = g - 18432;
    int which = i / 8192, e = i % 8192;
    int y = e / 64, rr = e % 64;
    int kx = (rr < 32) ? rr : rr - 64;  // signed frequency
    int m = ((kx * y) % 128 + 128) & 127;
    float th = PI2F * (float)m * (1.0f / 128.0f);
    t[g] = (which == 0) ? cosf(th) : (which == 1 ? sinf(th) : -sinf(th));
  } else if (g < 55296) {               // W2c / W2ns  [48 x 128]
    int i = g - 43008;
    int which = i / 6144, e = i % 6144;
    int ky = e / 128, x = e % 128;
    float v = 0.f;
    if (ky < 33) {
      float w = (ky == 0) ? 1.f : 2.f;  // hermitian doubling
      float th = PI2F * (float)((ky * x) & 127) * (1.0f / 128.0f);
      v = (which ? -w * sinf(th) : w * cosf(th)) * (1.0f / 16384.0f);
    }
    t[g] = v;
  }
}

// ---------------------------------------------------------------------------
// Kernel 1: 8-fold symmetrization of the 64x64 kernel fields.
// ---------------------------------------------------------------------------
__global__ void __launch_bounds__(256) symmetrize_kernel(
    const float* __restrict__ kin, float* __restrict__ ksym) {
  int fld = blockIdx.x;  // 0..127 = i*16 + j
  const float* x = kin + (long long)fld * 4096;
  float* o = ksym + (long long)fld * 4096;
  for (int e = threadIdx.x; e < 4096; e += 256) {
    int a = e >> 6, b = e & 63;
    int na = (64 - a) & 63, nb = (64 - b) & 63;
    float s = x[a * 64 + b] + x[nb * 64 + a] + x[na * 64 + nb] +
              x[b * 64 + na] + x[b * 64 + a] + x[a * 64 + nb] +
              x[nb * 64 + na] + x[na * 64 + b];
    o[e] = s * 0.125f;
  }
}

// ---------------------------------------------------------------------------
// Kernel 2: rfft2 of one 64x64 field -> [64 x 33] complex, as two WMMA GEMMs.
//   stage A: A[y,ky]  = f[y,:] . (cos | -sin)        M=64 K=64 N=48(pad)
//   stage B: F[kx,ky] = basis[kx,:] . (ReA | ImA)    M=64 K=64 N=48(pad)
// Field staged into LDS via GLOBAL_LOAD_ASYNC_TO_LDS_B32 (no VGPR roundtrip).
// ---------------------------------------------------------------------------
__global__ void __launch_bounds__(256) fwd_fft64_kernel(
    const float* __restrict__ in, float* __restrict__ outRe,
    float* __restrict__ outIm, const float* __restrict__ tab) {
  __shared__ float sF[64 * 65];   // field, padded stride (bank-conflict free)
  __shared__ float sRA[64 * 48];  // stage-A Re   (B-operand for stage B)
  __shared__ float sIA[64 * 48];  // stage-A Im

  const float* gcosA = tab + TAB_COSA;
  const float* gnsinA = tab + TAB_NSINA;
  const float* gCf = tab + TAB_CF;
  const float* gSf = tab + TAB_SF;
  const float* gNSf = tab + TAB_NSF;

  if (threadIdx.x == 0) {
    __builtin_prefetch(gcosA, 0, 3);
    __builtin_prefetch(gCf, 0, 3);
    __builtin_prefetch(gSf, 0, 3);
  }

  int fld = blockIdx.x;
  const float* src = in + (long long)fld * 4096;
  // async copy: 4096 dwords, global row stride 64 -> LDS row stride 65
#pragma unroll 4
  for (int e = threadIdx.x; e < 4096; e += 256) {
    unsigned ldsoff =
        (unsigned)(unsigned long long)(&sF[(e >> 6) * 65 + (e & 63)]);
    unsigned goff = (unsigned)e * 4u;
    asm volatile("global_load_async_to_lds_b32 %0, %1, %2"
                 :
                 : "v"(ldsoff), "v"(goff), "s"(src)
                 : "memory");
  }
  asm volatile("s_wait_asynccnt 0" ::: "memory");
  __syncthreads();

  int wave = threadIdx.x >> 5, lane = threadIdx.x & 31;
  int half = lane >> 4, r = lane & 15;

  // ---- stage A: 12 tiles x {Re,Im} = 24 wave-tasks
  for (int task = wave; task < 24; task += 8) {
    int which = task / 12;  // 0 -> ReA (cos), 1 -> ImA (-sin)
    int t = task % 12;
    int tm = t / 3, tn = t % 3;
    const float* Arow = sF + (tm * 16 + r) * 65;
    const float* Bcol = (which ? gnsinA : gcosA) + tn * 16 + r;
    v8f acc = mm_chain<64, 48>(Arow, Bcol, half, vzero8());
    float* dst = (which ? sIA : sRA) + (tm * 16) * 48 + tn * 16;
#pragma unroll
    for (int v = 0; v < 8; v++) dst[(v + 8 * half) * 48 + r] = acc[v];
  }
  __syncthreads();

  // ---- stage B: ReF = Cf@ReA + Sf@ImA ; ImF = Cf@ImA + (-Sf)@ReA
  for (int task = wave; task < 24; task += 8) {
    int which = task / 12;  // 0 -> ReF, 1 -> ImF
    int t = task % 12;
    int tm = t / 3, tn = t % 3;
    const float* Ap1 = gCf + (tm * 16 + r) * 64;
    const float* Bp1 = (which ? sIA : sRA) + tn * 16 + r;
    const float* Ap2 = (which ? gNSf : gSf) + (tm * 16 + r) * 64;
    const float* Bp2 = (which ? sRA : sIA) + tn * 16 + r;
    v8f acc = mm_chain<64, 48>(Ap1, Bp1, half, vzero8());
    acc = mm_chain<64, 48>(Ap2, Bp2, half, acc);
    int ky = tn * 16 + r;
    if (ky < 33) {
      float* o = (which ? outIm : outRe) + (long long)fld * 2112;
#pragma unroll
      for (int v = 0; v < 8; v++) o[(tm * 16 + v + 8 * half) * 33 + ky] = acc[v];
    }
  }
}

// ---------------------------------------------------------------------------
// Kernel 3: per-frequency complex channel contraction via WMMA.
//   conv[b,j] = sum_i F[b,i] * K[i,j]   (M=16 b, N=16 j, K=8 i), 2112 points.
//   One wave per frequency point; 8 WMMAs each.
// ---------------------------------------------------------------------------
__global__ void __launch_bounds__(256) einsum_kernel(
    const float* __restrict__ ReF, const float* __restrict__ ImF,
    const float* __restrict__ ReK, const float* __restrict__ ImK,
    float* __restrict__ ReC, float* __restrict__ ImC) {
  int wave = threadIdx.x >> 5, lane = threadIdx.x & 31;
  int half = lane >> 4, r = lane & 15;
  int p = blockIdx.x * 8 + wave;  // 0..2111

  v8f accRR = vzero8(), accII = vzero8(), accIm = vzero8();
#pragma unroll
  for (int kk = 0; kk < 8; kk += 4) {
    int k0 = kk + 2 * half, k1 = k0 + 1;
    v2f aR, aI, bR, bI;
    aR.x = ReF[(long long)(r * 8 + k0) * 2112 + p];
    aR.y = ReF[(long long)(r * 8 + k1) * 2112 + p];
    aI.x = ImF[(long long)(r * 8 + k0) * 2112 + p];
    aI.y = ImF[(long long)(r * 8 + k1) * 2112 + p];
    bR.x = ReK[(long long)(k0 * 16 + r) * 2112 + p];
    bR.y = ReK[(long long)(k1 * 16 + r) * 2112 + p];
    bI.x = ImK[(long long)(k0 * 16 + r) * 2112 + p];
    bI.y = ImK[(long long)(k1 * 16 + r) * 2112 + p];
    accRR = wmma4(aR, bR, accRR);
    accII = wmma4(aI, bI, accII);
    accIm = wmma4(aR, bI, accIm);
    accIm = wmma4(aI, bR, accIm);
  }
  v8f re = accRR - accII;
#pragma unroll
  for (int v = 0; v < 8; v++) {
    long long fo = (long long)((v + 8 * half) * 16 + r) * 2112 + p;
    ReC[fo] = re[v];
    ImC[fo] = accIm[v];
  }
}

// ---------------------------------------------------------------------------
// Kernel 4: fused upsample(+uncurl) + irfft2 -> 128x128 real field.
//   Each block computes one 64-row half of one output field.
//   stage 1: T[y,ky] = sum_kx G[kx,ky] e^{+2pi i kx y/128}   (M=64 K=64 N=48)
//   stage 2: out[y,x] = ReT@W2c + ImT@W2ns                   (M=64 K=48 N=128)
//   mode: 0 = plain*2 (fr), 1 = * i*2*ky/|k|^2 (u), 2 = * -i*2*kx/|k|^2 (v)
// ---------------------------------------------------------------------------
__global__ void __launch_bounds__(256) ifft2_kernel(
    const float* __restrict__ gRe, const float* __restrict__ gIm,
    float* __restrict__ out, const float* __restrict__ tab, int mode, int dmod,
    long long s1, long long s2) {
  __shared__ float sGR[64 * 48];  // spectrum Re (B-operand, stage 1)
  __shared__ float sGI[64 * 48];
  __shared__ float sTR[64 * 49];  // stage-1 Re (A-operand, stage 2), padded
  __shared__ float sTI[64 * 49];

  const float* E1c = tab + TAB_E1C;
  const float* E1s = tab + TAB_E1S;
  const float* NE1s = tab + TAB_NE1S;
  const float* W2c = tab + TAB_W2C;
  const float* W2ns = tab + TAB_W2NS;

  if (threadIdx.x == 0) {
    __builtin_prefetch(E1c, 0, 3);
    __builtin_prefetch(W2c, 0, 3);
    __builtin_prefetch(W2ns, 0, 3);
  }

  int blk = blockIdx.x;
  int fld = blk >> 1;
  int hfy = blk & 1;  // output row half

  // load spectrum with upsample zeroing + uncurl factor, zero-pad ky to 48
  const float* pR = gRe + (long long)fld * 2112;
  const float* pI = gIm + (long long)fld * 2112;
  for (int e = threadIdx.x; e < 3072; e += 256) {
    int rr = e / 48, c = e % 48;
    float greo = 0.f, gimo = 0.f;
    if (c < 32 && rr != 32) {  // drop nyquist row/col; c>=32 padded zero
      float re = pR[rr * 33 + c], im = pI[rr * 33 + c];
      if (mode == 0) {
        greo = 2.f * re;
        gimo = 2.f * im;
      } else {
        float kx = (rr < 32) ? (float)rr : (float)(rr - 64);
        float ky = (float)c;
        float den = kx * kx + ky * ky;
        if (den == 0.f) den = 1.f;
        if (mode == 1) {  // * i*ky/den * 2
          float s = 2.f * ky / den;
          greo = -im * s;
          gimo = re * s;
        } else {          // * -i*kx/den * 2
          float s = 2.f * kx / den;
          greo = im * s;
          gimo = -re * s;
        }
      }
    }
    sGR[e] = greo;
    sGI[e] = gimo;
  }
  __syncthreads();

  int wave = threadIdx.x >> 5, lane = threadIdx.x & 31;
  int half = lane >> 4, r = lane & 15;

  // ---- stage 1: ReT = E1c@GR + (-E1s)@GI ; ImT = E1s@GR + E1c@GI
  for (int task = wave; task < 24; task += 8) {
    int which = task / 12;  // 0 -> ReT, 1 -> ImT
    int t = task % 12;
    int tm = t / 3, tn = t % 3;
    int y = hfy * 64 + tm * 16 + r;
    const float* Ap1 = (which ? E1s : E1c) + y * 64;
    const float* Ap2 = (which ? E1c : NE1s) + y * 64;
    const float* Bp1 = sGR + tn * 16 + r;
    const float* Bp2 = sGI + tn * 16 + r;
    v8f acc = mm_chain<64, 48>(Ap1, Bp1, half, vzero8());
    acc = mm_chain<64, 48>(Ap2, Bp2, half, acc);
    float* dst = (which ? sTI : sTR) + (tm * 16) * 49 + tn * 16;
#pragma unroll
    for (int v = 0; v < 8; v++) dst[(v + 8 * half) * 49 + r] = acc[v];
  }
  __syncthreads();

  // ---- stage 2: out[y,x] = ReT@W2c + ImT@W2ns  (K=48, zero-padded tail)
  long long obase =
      (long long)(fld / dmod) * s1 + (long long)(fld % dmod) * s2 +
      (long long)hfy * 64 * 128;
  float* o = out + obase;
  for (int task = wave; task < 32; task += 8) {
    int tm = task / 8, tn = task % 8;
    const float* A1 = sTR + (tm * 16 + r) * 49;
    const float* A2 = sTI + (tm * 16 + r) * 49;
    const float* B1 = W2c + tn * 16 + r;
    const float* B2 = W2ns + tn * 16 + r;
    v8f acc = mm_chain<48, 128>(A1, B1, half, vzero8());
    acc = mm_chain<48, 128>(A2, B2, half, acc);
#pragma unroll
    for (int v = 0; v < 8; v++)
      o[(tm * 16 + v + 8 * half) * 128 + tn * 16 + r] = acc[v];
  }
}

// ---------------------------------------------------------------------------
// Kernel 5: antisymmetric cross products -> output channels 8..127.
// ---------------------------------------------------------------------------
__global__ void __launch_bounds__(256) cross_kernel(
    const float* __restrict__ u, const float* __restrict__ v,
    float* __restrict__ out) {
  const long long total = 16LL * 120 * 16384;
  long long base = ((long long)blockIdx.x * 256 + threadIdx.x) * 4;
#pragma unroll
  for (int q = 0; q < 4; q++) {
    long long e = base + q;
    if (e >= total) return;
    int x = (int)(e & 16383);
    long long rest = e >> 14;
    int pair = (int)(rest % 120);
    int b = (int)(rest / 120);
    int i = 0, rem = pair;
    while (rem >= 15 - i) { rem -= 15 - i; ++i; }
    int j = i + 1 + rem;
    const float* ub = u + (long long)b * 16 * 16384;
    const float* vb = v + (long long)b * 16 * 16384;
    float ui = ub[i * 16384 + x], uj = ub[j * 16384 + x];
    float vi = vb[i * 16384 + x], vj = vb[j * 16384 + x];
    out[(long long)(b * 128 + 8 + pair) * 16384 + x] = ui * vj - uj * vi;
  }
}

// ---------------------------------------------------------------------------
extern "C" void kernel_launch(void* const* d_in, const int* in_sizes, int n_in,
                              void* d_out, int out_size, void* d_ws,
                              size_t ws_size, hipStream_t stream) {
  const float* f = (const float*)d_in[0];    // [16,8,64,64]
  const float* kin = (const float*)d_in[1];  // [1,8,16,64,64]
  float* out = (float*)d_out;                // [16,128,128,128]
  float* ws = (float*)d_ws;

  float* tab = ws;
  float* ReFf = ws + WS_REFF;
  float* ImFf = ws + WS_IMFF;
  float* ksym = ws + WS_KSYM;
  float* ReKs = ws + WS_REKS;
  float* ImKs = ws + WS_IMKS;
  float* ReC = ws + WS_RECONV;
  float* ImC = ws + WS_IMCONV;
  float* ubuf = ws + WS_U;
  float* vbuf = ws + WS_V;

  build_tables_kernel<<<216, 256, 0, stream>>>(tab);
  symmetrize_kernel<<<128, 256, 0, stream>>>(kin, ksym);
  fwd_fft64_kernel<<<128, 256, 0, stream>>>(f, ReFf, ImFf, tab);
  fwd_fft64_kernel<<<128, 256, 0, stream>>>(ksym, ReKs, ImKs, tab);
  einsum_kernel<<<264, 256, 0, stream>>>(ReFf, ImFf, ReKs, ImKs, ReC, ImC);
  // u, v: 256 fields x 2 halves; contiguous in ws
  ifft2_kernel<<<512, 256, 0, stream>>>(ReC, ImC, ubuf, tab, 1, 16,
                                        16LL * 16384, 16384LL);
  ifft2_kernel<<<512, 256, 0, stream>>>(ReC, ImC, vbuf, tab, 2, 16,
                                        16LL * 16384, 16384LL);
  // fr: 128 fields (b,i) -> out channel b*128 + i
  ifft2_kernel<<<256, 256, 0, stream>>>(ReFf, ImFf, out, tab, 0, 8,
                                        128LL * 16384, 16384LL);
  cross_kernel<<<30720, 256, 0, stream>>>(ubuf, vbuf, out);
}